// DGCNN_33672543600663
// MI455X (gfx1250) — compile-verified
//
#include <hip/hip_runtime.h>
#include <math.h>

// ---------------------------------------------------------------------------
// DGCNN-style batched GCN for MI455X (gfx1250, wave32).
//   t = h @ W          -> V_WMMA_F32_16X16X4_F32 (fp32 matrix pipe, exact math)
//   scatter-add        -> wave-per-edge, lane == channel (LAT=32 == wave32),
//                         global_atomic_add_f32 (hardware FP atomics)
//   Memory-bound: ~5 GB of gather/scatter traffic dominates (HBM 23.3 TB/s).
// ---------------------------------------------------------------------------

typedef __attribute__((ext_vector_type(2))) float v2f;
typedef __attribute__((ext_vector_type(8))) float v8f;

#define IN_DIM 128
#define LAT    32
#define HID    128

// ---------------------------------------------------------------------------
// Degree / normalization
// ---------------------------------------------------------------------------
__global__ __launch_bounds__(256) void k_init_deg(int* deg, int n) {
    int i = blockIdx.x * blockDim.x + threadIdx.x;
    if (i < n) deg[i] = 1;                    // self-loop
}

__global__ __launch_bounds__(256) void k_count_deg(const int* __restrict__ dst,
                                                   int* deg, int e) {
    int i = blockIdx.x * blockDim.x + threadIdx.x;
    if (i < e) atomicAdd(&deg[dst[i]], 1);
}

__global__ __launch_bounds__(256) void k_dinv(const int* __restrict__ deg,
                                              float* __restrict__ dinv, int n) {
    int i = blockIdx.x * blockDim.x + threadIdx.x;
    if (i < n) dinv[i] = rsqrtf((float)deg[i]);   // deg >= 1 always
}

// ---------------------------------------------------------------------------
// WMMA GEMM: T = H(K cols) @ W(K x 32), fused self-loop init Agg = T * dinv^2
// Block = 128 threads = 4 waves; each wave owns a 16x32 output tile (64 rows
// per block). LDS-staged, padded to kill bank conflicts.
//
// f32 WMMA fragment layouts (ISA 7.12.2):
//   A 16x4 : lane<16 -> (M=lane, K=k0+{0,1});  lane>=16 -> (M=lane-16, K=k0+{2,3})
//   B 4x16 : vgpr v, lanes0-15 -> row K=k0+v ; lanes16-31 -> row K=k0+v+2
//   C/D    : vgpr v, lanes0-15 -> M=v ; lanes16-31 -> M=v+8 ; N=lane&15
// ---------------------------------------------------------------------------
template <int K>
__global__ __launch_bounds__(128) void gemm_xform(const float* __restrict__ H,
                                                  const float* __restrict__ W,
                                                  const float* __restrict__ dinv,
                                                  float* __restrict__ T,
                                                  float* __restrict__ Agg,
                                                  int nrows) {
    constexpr int KP = K + 1;          // pad row stride -> conflict-free A loads
    __shared__ float sH[64 * KP];
    __shared__ float sW[K * 33];

    const int tid = threadIdx.x;
    const int rowBlock = blockIdx.x * 64;

    // stage W (K x 32), padded stride 33
    for (int i = tid; i < K * 32; i += 128) {
        int k = i >> 5, n = i & 31;
        sW[k * 33 + n] = W[i];
    }
    // stage H tile (64 x K), zero-fill OOB rows so EXEC can stay all-ones
    for (int i = tid; i < 64 * K; i += 128) {
        int r = i / K, c = i % K;      // K is a power of two
        int gr = rowBlock + r;
        sH[r * KP + c] = (gr < nrows) ? H[(size_t)gr * K + c] : 0.0f;
    }
    __syncthreads();

    const int wave = tid >> 5;
    const int lane = tid & 31;
    const int m    = lane & 15;
    const int half = lane >> 4;

    v8f acc0 = {};   // output cols 0..15
    v8f acc1 = {};   // output cols 16..31
    const float* sHrow = &sH[(wave * 16 + m) * KP];

#pragma unroll 4
    for (int k0 = 0; k0 < K; k0 += 4) {
        const int ka = k0 + 2 * half;
        v2f a;  a.x  = sHrow[ka];               a.y  = sHrow[ka + 1];
        v2f b0; b0.x = sW[ka * 33 + m];         b0.y = sW[(ka + 1) * 33 + m];
        v2f b1; b1.x = sW[ka * 33 + 16 + m];    b1.y = sW[(ka + 1) * 33 + 16 + m];
        acc0 = __builtin_amdgcn_wmma_f32_16x16x4_f32(false, a, false, b0,
                                                     (short)0, acc0, false, false);
        acc1 = __builtin_amdgcn_wmma_f32_16x16x4_f32(false, a, false, b1,
                                                     (short)0, acc1, false, false);
    }

    const int rowBase = rowBlock + wave * 16;
#pragma unroll
    for (int v = 0; v < 8; ++v) {
        int r = rowBase + v + 8 * half;
        if (r < nrows) {
            float d0 = acc0[v], d1 = acc1[v];
            size_t o = (size_t)r * LAT;
            T[o + m]        = d0;
            T[o + 16 + m]   = d1;
            float di = dinv[r];
            float s  = di * di;                 // self-loop norm dinv[i]^2
            Agg[o + m]      = d0 * s;
            Agg[o + 16 + m] = d1 * s;
        }
    }
}

// ---------------------------------------------------------------------------
// Edge scatter: one wave per edge, lane == channel. Agg[dst] += T[src]*norm.
// ---------------------------------------------------------------------------
__global__ __launch_bounds__(256) void k_scatter(const int* __restrict__ src,
                                                 const int* __restrict__ dst,
                                                 const float* __restrict__ dinv,
                                                 const float* __restrict__ T,
                                                 float* __restrict__ Agg, int e) {
    int gid = blockIdx.x * blockDim.x + threadIdx.x;
    int ed  = gid >> 5;
    int c   = gid & 31;
    if (ed >= e) return;
    int s = src[ed];
    int d = dst[ed];
    float nrm = dinv[s] * dinv[d];
    float v = T[(size_t)s * LAT + c] * nrm;
    unsafeAtomicAdd(&Agg[(size_t)d * LAT + c], v);   // global_atomic_add_f32
}

// ---------------------------------------------------------------------------
// H = tanh(Agg + b), in place
// ---------------------------------------------------------------------------
__global__ __launch_bounds__(256) void k_finish(float* __restrict__ h,
                                                const float* __restrict__ b,
                                                int total) {
    int i = blockIdx.x * blockDim.x + threadIdx.x;
    if (i < total) h[i] = tanhf(h[i] + b[i & (LAT - 1)]);
}

// ---------------------------------------------------------------------------
// Readout: idx[g] = searchsorted(batch, g) (left); pooled = concat(h1,h2,h3)[idx]
// Block per graph, 96 threads.
// ---------------------------------------------------------------------------
__global__ __launch_bounds__(96) void k_pool(const int* __restrict__ batch,
                                             const float* __restrict__ h1,
                                             const float* __restrict__ h2,
                                             const float* __restrict__ h3,
                                             float* __restrict__ pooled, int n) {
    int g = blockIdx.x;
    int lo = 0, hi = n;
    while (lo < hi) {                       // first i with batch[i] >= g
        int mid = (lo + hi) >> 1;
        if (batch[mid] < g) lo = mid + 1; else hi = mid;
    }
    int idx = lo;
    int c = threadIdx.x;
    float v;
    if (c < 32)       v = h1[(size_t)idx * LAT + c];
    else if (c < 64)  v = h2[(size_t)idx * LAT + (c - 32)];
    else              v = h3[(size_t)idx * LAT + (c - 64)];
    pooled[(size_t)g * 96 + c] = v;
}

// ---------------------------------------------------------------------------
// MLP head: hidden = relu(pooled @ Wl1 + bl1); block per graph, 128 threads
// ---------------------------------------------------------------------------
__global__ __launch_bounds__(128) void k_mlp1(const float* __restrict__ pooled,
                                              const float* __restrict__ Wl1,
                                              const float* __restrict__ bl1,
                                              float* __restrict__ hidden) {
    __shared__ float sp[96];
    int g = blockIdx.x;
    int o = threadIdx.x;
    if (o < 96) sp[o] = pooled[(size_t)g * 96 + o];
    __syncthreads();
    float acc = bl1[o];
#pragma unroll 8
    for (int k = 0; k < 96; ++k) acc += sp[k] * Wl1[k * HID + o];
    hidden[(size_t)g * HID + o] = fmaxf(acc, 0.0f);
}

// logits = log_softmax(hidden @ Wl2 + bl2); one thread per graph
__global__ __launch_bounds__(256) void k_mlp2(const float* __restrict__ hidden,
                                              const float* __restrict__ Wl2,
                                              const float* __restrict__ bl2,
                                              float* __restrict__ out, int g_total) {
    int g = blockIdx.x * blockDim.x + threadIdx.x;
    if (g >= g_total) return;
    float l0 = bl2[0], l1 = bl2[1];
    const float* hrow = &hidden[(size_t)g * HID];
#pragma unroll 8
    for (int k = 0; k < HID; ++k) {
        float h = hrow[k];
        l0 += h * Wl2[2 * k];
        l1 += h * Wl2[2 * k + 1];
    }
    float m   = fmaxf(l0, l1);
    float lse = m + logf(expf(l0 - m) + expf(l1 - m));
    out[2 * g]     = l0 - lse;
    out[2 * g + 1] = l1 - lse;
}

// ---------------------------------------------------------------------------
extern "C" void kernel_launch(void* const* d_in, const int* in_sizes, int n_in,
                              void* d_out, int out_size, void* d_ws, size_t ws_size,
                              hipStream_t stream) {
    const float* x     = (const float*)d_in[0];
    const int*   ei    = (const int*)d_in[1];
    const int*   batch = (const int*)d_in[2];
    const float* W1    = (const float*)d_in[3];
    const float* b1    = (const float*)d_in[4];
    const float* W2    = (const float*)d_in[5];
    const float* b2    = (const float*)d_in[6];
    const float* W3    = (const float*)d_in[7];
    const float* b3    = (const float*)d_in[8];
    const float* Wl1   = (const float*)d_in[9];
    const float* bl1   = (const float*)d_in[10];
    const float* Wl2   = (const float*)d_in[11];
    const float* bl2   = (const float*)d_in[12];
    float* out = (float*)d_out;

    const int N = in_sizes[0] / IN_DIM;
    const int E = in_sizes[1] / 2;
    const int G = out_size / 2;
    const int* src = ei;
    const int* dst = ei + E;

    // Workspace carve-up (all 256B aligned). Total ~106 MB.
    char* ws = (char*)d_ws;
    size_t off = 0;
    auto alloc = [&](size_t bytes) {
        void* p = ws + off;
        off += (bytes + 255) & ~(size_t)255;
        return p;
    };
    int*   deg    = (int*)  alloc((size_t)N * sizeof(int));
    float* dinv   = (float*)alloc((size_t)N * sizeof(float));
    float* T      = (float*)alloc((size_t)N * LAT * sizeof(float));
    float* h1     = (float*)alloc((size_t)N * LAT * sizeof(float));
    float* h2     = (float*)alloc((size_t)N * LAT * sizeof(float));
    float* h3     = (float*)alloc((size_t)N * LAT * sizeof(float));
    float* pooled = (float*)alloc((size_t)G * 96 * sizeof(float));
    float* hidden = (float*)alloc((size_t)G * HID * sizeof(float));
    (void)ws_size; (void)n_in;

    const int gemmBlocks = (N + 63) / 64;
    const int elemBlocks = ((N * LAT) + 255) / 256;
    const int edgeBlocks = (int)(((long long)E * 32 + 255) / 256);

    // degree + normalization
    k_init_deg <<<(N + 255) / 256, 256, 0, stream>>>(deg, N);
    k_count_deg<<<(E + 255) / 256, 256, 0, stream>>>(dst, deg, E);
    k_dinv     <<<(N + 255) / 256, 256, 0, stream>>>(deg, dinv, N);

    // layer 1 (K = 128)
    gemm_xform<IN_DIM><<<gemmBlocks, 128, 0, stream>>>(x, W1, dinv, T, h1, N);
    k_scatter<<<edgeBlocks, 256, 0, stream>>>(src, dst, dinv, T, h1, E);
    k_finish <<<elemBlocks, 256, 0, stream>>>(h1, b1, N * LAT);

    // layer 2 (K = 32)
    gemm_xform<LAT><<<gemmBlocks, 128, 0, stream>>>(h1, W2, dinv, T, h2, N);
    k_scatter<<<edgeBlocks, 256, 0, stream>>>(src, dst, dinv, T, h2, E);
    k_finish <<<elemBlocks, 256, 0, stream>>>(h2, b2, N * LAT);

    // layer 3 (K = 32)
    gemm_xform<LAT><<<gemmBlocks, 128, 0, stream>>>(h2, W3, dinv, T, h3, N);
    k_scatter<<<edgeBlocks, 256, 0, stream>>>(src, dst, dinv, T, h3, E);
    k_finish <<<elemBlocks, 256, 0, stream>>>(h3, b3, N * LAT);

    // readout + MLP head
    k_pool<<<G, 96, 0, stream>>>(batch, h1, h2, h3, pooled, N);
    k_mlp1<<<G, 128, 0, stream>>>(pooled, Wl1, bl1, hidden);
    k_mlp2<<<(G + 255) / 256, 256, 0, stream>>>(hidden, Wl2, bl2, out, G);
}